// GumbelVQ_25589415150078
// MI455X (gfx1250) — compile-verified
//
#include <hip/hip_runtime.h>
#include <cstdint>

// ---------------------------------------------------------------------------
// GumbelVQ forward: idx = argmax_k( (z_flat @ proj_w^T)[n,k] + gumbel[n,k] )
// z_q = embedding[idx] ; commit = 0.25*mean((z_flat - embedding[idx])^2)
// perplexity/usage from histogram of idx.
//
// N = K = 8192, D = 8, z_e is (8,8,32,32) NCHW, z_flat[n][d] row-major NHWC.
// Logits tile 16x16 computed with two chained V_WMMA_F32_16X16X4_F32.
// JAX threefry2x32 with key (0,42); element i<2^25 -> out0, i>=2^25 -> out1,
// so rows n and n+4096 share one threefry block (processed together).
// ---------------------------------------------------------------------------

typedef __attribute__((ext_vector_type(2))) float v2f;
typedef __attribute__((ext_vector_type(8))) float v8f;

#define NROWS 8192
#define NCODES 8192

__device__ __forceinline__ uint32_t rotl32(uint32_t x, int r) {
  return (x << r) | (x >> (32 - r));
}

// Threefry-2x32-20 with key (0, 42) -- exact JAX PRNG for jax.random.key(42)
__device__ __forceinline__ void threefry2x32_42(uint32_t x0, uint32_t x1,
                                                uint32_t& o0, uint32_t& o1) {
  const uint32_t ks0 = 0u;
  const uint32_t ks1 = 42u;
  const uint32_t ks2 = 0x1BD11BDAu ^ 42u;
  x0 += ks0; x1 += ks1;
#define TF_R4(a, b, c, d)                       \
  x0 += x1; x1 = rotl32(x1, a); x1 ^= x0;       \
  x0 += x1; x1 = rotl32(x1, b); x1 ^= x0;       \
  x0 += x1; x1 = rotl32(x1, c); x1 ^= x0;       \
  x0 += x1; x1 = rotl32(x1, d); x1 ^= x0;
  TF_R4(13, 15, 26, 6);  x0 += ks1; x1 += ks2 + 1u;
  TF_R4(17, 29, 16, 24); x0 += ks2; x1 += ks0 + 2u;
  TF_R4(13, 15, 26, 6);  x0 += ks0; x1 += ks1 + 3u;
  TF_R4(17, 29, 16, 24); x0 += ks1; x1 += ks2 + 4u;
  TF_R4(13, 15, 26, 6);  x0 += ks2; x1 += ks0 + 5u;
#undef TF_R4
  o0 = x0; o1 = x1;
}

// jax.random.uniform mantissa trick + minval=1e-20 clamp
__device__ __forceinline__ float jax_unit(uint32_t bits) {
  float f = __uint_as_float((bits >> 9) | 0x3f800000u) - 1.0f;
  return fmaxf(1e-20f, f + 1e-20f);
}

__device__ __forceinline__ float gumbel_of(float u) {
  return -logf(-logf(u));
}

// Monotone (value, index) packing: larger value wins; tie -> lower index wins.
__device__ __forceinline__ unsigned long long pack_key(float v, uint32_t idx) {
  uint32_t u = __float_as_uint(v);
  u ^= (u & 0x80000000u) ? 0xFFFFFFFFu : 0x80000000u;
  return ((unsigned long long)u << 32) | (uint32_t)(~idx);
}

// ---------------------------------------------------------------------------
// Phase 0: zero scratch (packed keys + counts + commit accumulator)
// ---------------------------------------------------------------------------
__global__ void vq_zero(uint32_t* __restrict__ ws_words, int nwords) {
  int i = blockIdx.x * blockDim.x + threadIdx.x;
  if (i < nwords) ws_words[i] = 0u;
}

// ---------------------------------------------------------------------------
// Phase 1: WMMA logits + gumbel + online argmax.
// 512 blocks x 128 threads = 2048 waves.
// wave w: row-pair tile rt = w & 255 (rows rt*16..+15 and +4096),
//         column chunk  = w >> 8  (col tiles chunk*64 .. +63)
// ---------------------------------------------------------------------------
__global__ __launch_bounds__(128) void vq_phase1(
    const float* __restrict__ z_e, const float* __restrict__ proj_w,
    unsigned long long* __restrict__ packed) {
  const int waveInBlk = threadIdx.x >> 5;
  const int lane = threadIdx.x & 31;
  const int w = blockIdx.x * 4 + waveInBlk;  // 0..2047
  const int rt = w & 255;
  const int chunk = w >> 8;
  const int half = lane >> 4;   // 0/1
  const int nn = lane & 15;

  // ---- A fragments (16x4 f32 layout: lane holds row M=nn; VGPR0=K even,
  //      VGPR1=K odd, K base = 2*half) for rows n0 and n0+4096.
  const int kb = half * 2;
  const int n0 = rt * 16 + nn;
  const int n1 = n0 + 4096;
  auto zf = [&](int n, int d) {
    // z_flat[n][d] = z_e[(n>>10)*8192 + d*1024 + (n&1023)]
    return z_e[(n >> 10) * 8192 + d * 1024 + (n & 1023)];
  };
  v2f a0lo, a0hi, a1lo, a1hi;
  a0lo.x = zf(n0, kb);     a0lo.y = zf(n0, kb + 1);
  a0hi.x = zf(n0, kb + 4); a0hi.y = zf(n0, kb + 5);
  a1lo.x = zf(n1, kb);     a1lo.y = zf(n1, kb + 1);
  a1hi.x = zf(n1, kb + 4); a1hi.y = zf(n1, kb + 5);

  float m0[8], m1[8];
  uint32_t id0[8], id1[8];
#pragma unroll
  for (int r = 0; r < 8; ++r) {
    m0[r] = -3.4e38f; m1[r] = -3.4e38f; id0[r] = 0u; id1[r] = 0u;
  }

  const int ct0 = chunk * 64;
  for (int ct = ct0; ct < ct0 + 64; ++ct) {
    // ---- B fragment (4x16 f32: lane holds col N=nn; VGPR0=K even,
    //      VGPR1=K odd, K base = 2*half); proj_w is (K,D) row-major.
    const int ncol = ct * 16 + nn;
    const float* bp = proj_w + ncol * 8 + kb;
    v2f blo, bhi;
    blo.x = bp[0]; blo.y = bp[1];
    bhi.x = bp[4]; bhi.y = bp[5];

    v8f c0 = {};
    c0 = __builtin_amdgcn_wmma_f32_16x16x4_f32(false, a0lo, false, blo,
                                               (short)0, c0, false, false);
    c0 = __builtin_amdgcn_wmma_f32_16x16x4_f32(false, a0hi, false, bhi,
                                               (short)0, c0, false, false);
    v8f c1 = {};
    c1 = __builtin_amdgcn_wmma_f32_16x16x4_f32(false, a1lo, false, blo,
                                               (short)0, c1, false, false);
    c1 = __builtin_amdgcn_wmma_f32_16x16x4_f32(false, a1hi, false, bhi,
                                               (short)0, c1, false, false);

    const uint32_t k = (uint32_t)ncol;
#pragma unroll
    for (int r = 0; r < 8; ++r) {
      // C layout: lane, VGPR r -> (M = r + 8*half, N = nn)
      const uint32_t row0 = (uint32_t)(rt * 16 + r + 8 * half);
      const uint32_t i0 = row0 * 8192u + k;  // < 2^25 since row0 < 4096
      uint32_t o0, o1;
      threefry2x32_42(i0, i0 + 0x2000000u, o0, o1);
      const float x0 = c0[r] + gumbel_of(jax_unit(o0));
      const float x1 = c1[r] + gumbel_of(jax_unit(o1));
      if (x0 > m0[r]) { m0[r] = x0; id0[r] = k; }
      if (x1 > m1[r]) { m1[r] = x1; id1[r] = k; }
    }
  }

  // ---- reduce across the 16 lanes sharing each row (xor stays in half)
#pragma unroll
  for (int r = 0; r < 8; ++r) {
#pragma unroll
    for (int s = 1; s < 16; s <<= 1) {
      float om = __shfl_xor(m0[r], s, 32);
      uint32_t oi = (uint32_t)__shfl_xor((int)id0[r], s, 32);
      if (om > m0[r] || (om == m0[r] && oi < id0[r])) { m0[r] = om; id0[r] = oi; }
      om = __shfl_xor(m1[r], s, 32);
      oi = (uint32_t)__shfl_xor((int)id1[r], s, 32);
      if (om > m1[r] || (om == m1[r] && oi < id1[r])) { m1[r] = om; id1[r] = oi; }
    }
  }

  if (nn == 0) {
#pragma unroll
    for (int r = 0; r < 8; ++r) {
      const int row0 = rt * 16 + r + 8 * half;
      atomicMax(&packed[row0], pack_key(m0[r], id0[r]));
      atomicMax(&packed[row0 + 4096], pack_key(m1[r], id1[r]));
    }
  }
}

// ---------------------------------------------------------------------------
// Phase 2: decode idx, gather embedding, commit-loss partial, histogram.
// 32 blocks x 256 threads = one thread per row.
// ---------------------------------------------------------------------------
__global__ __launch_bounds__(256) void vq_phase2(
    const float* __restrict__ z_e, const float* __restrict__ embedding,
    const unsigned long long* __restrict__ packed, float* __restrict__ out,
    float* __restrict__ counts, float* __restrict__ commit_acc) {
  const int n = blockIdx.x * 256 + threadIdx.x;  // 0..8191
  const unsigned long long p = packed[n];
  const uint32_t idx = ~(uint32_t)(p & 0xFFFFFFFFull);
  out[65536 + n] = (float)idx;  // indices output (numeric cast)
  const int b = n >> 10, hw = n & 1023;
  float sq = 0.0f;
#pragma unroll
  for (int d = 0; d < 8; ++d) {
    const float e = embedding[idx * 8u + (uint32_t)d];
    const int off = b * 8192 + d * 1024 + hw;  // NCHW position for (n, d)
    out[off] = e;                              // z_q = embedding[idx]
    const float z = z_e[off];
    const float t = z - e;
    sq += t * t;
  }
  atomicAdd(commit_acc, sq);
  atomicAdd(&counts[idx], 1.0f);
}

// ---------------------------------------------------------------------------
// Phase 3: scalar outputs (commit loss, perplexity, usage).
// ---------------------------------------------------------------------------
__global__ __launch_bounds__(256) void vq_phase3(
    const float* __restrict__ counts, const float* __restrict__ commit_acc,
    float* __restrict__ out) {
  __shared__ float s_p[256];
  __shared__ float s_n[256];
  float plogp = 0.0f, nz = 0.0f;
  for (int k = threadIdx.x; k < NCODES; k += 256) {
    const float c = counts[k];
    const float avg = c * (1.0f / 8192.0f);
    plogp += avg * logf(avg + 1e-10f);
    nz += (c > 0.0f) ? 1.0f : 0.0f;
  }
  s_p[threadIdx.x] = plogp;
  s_n[threadIdx.x] = nz;
  __syncthreads();
  for (int s = 128; s > 0; s >>= 1) {
    if ((int)threadIdx.x < s) {
      s_p[threadIdx.x] += s_p[threadIdx.x + s];
      s_n[threadIdx.x] += s_n[threadIdx.x + s];
    }
    __syncthreads();
  }
  if (threadIdx.x == 0) {
    out[73728] = 0.25f * commit_acc[0] * (1.0f / 65536.0f);  // BETA * mean
    out[73729] = expf(-s_p[0]);                              // perplexity
    out[73730] = s_n[0] * (1.0f / 8192.0f);                  // usage
  }
}

// ---------------------------------------------------------------------------
extern "C" void kernel_launch(void* const* d_in, const int* in_sizes, int n_in,
                              void* d_out, int out_size, void* d_ws,
                              size_t ws_size, hipStream_t stream) {
  const float* z_e = (const float*)d_in[0];        // (8,8,32,32)
  const float* embedding = (const float*)d_in[1];  // (8192,8)
  const float* proj_w = (const float*)d_in[2];     // (8192,8)
  float* out = (float*)d_out;

  // ws layout: [0,64KB) packed u64 keys; [64KB,96KB) counts f32; then 1 f32.
  unsigned long long* packed = (unsigned long long*)d_ws;
  float* counts = (float*)((char*)d_ws + (size_t)NROWS * 8);
  float* commit_acc = counts + NCODES;

  const int zero_words = NROWS * 2 + NCODES + 1;  // 24577
  vq_zero<<<(zero_words + 255) / 256, 256, 0, stream>>>((uint32_t*)d_ws,
                                                        zero_words);
  vq_phase1<<<512, 128, 0, stream>>>(z_e, proj_w, packed);
  vq_phase2<<<32, 256, 0, stream>>>(z_e, embedding, packed, out, counts,
                                    commit_acc);
  vq_phase3<<<1, 256, 0, stream>>>(counts, commit_acc, out);
}